// CensoredLoss_73023033967187
// MI455X (gfx1250) — compile-verified
//
#include <hip/hip_runtime.h>
#include <cstdint>
#include <cstddef>

#define T_INTERVALS 512
#define VW 5           // targets width (slot 0 = censor)
#define OW 4           // outputs width (VW-1)
#define TPB 256
#define LOSS_EPS 1e-8f

// Low 32 bits of a generic pointer to LDS are the LDS byte offset
// (ISA 10.2: LDS aperture in addr[63:32], offset in addr[31:0]).
__device__ __forceinline__ unsigned lds_byte_off(const void* p) {
    return (unsigned)(unsigned long long)p;
}

__global__ __launch_bounds__(TPB) void censored_loss_rows(
    const float* __restrict__ outputs,   // [B, T, 4]
    const float* __restrict__ targets,   // [B, T, 5]
    float* __restrict__ acc,             // partials[2*B] (mode 0) or acc[2] (mode 1)
    int use_atomic)
{
    __shared__ __align__(16) float sT[T_INTERVALS * VW];   // 2560 floats = 10240 B
    __shared__ __align__(16) float sO[T_INTERVALS * OW];   // 2048 floats =  8192 B
    __shared__ int   sI[TPB];
    __shared__ float sF[TPB];

    const int b   = blockIdx.x;
    const int tid = threadIdx.x;

    const float* gT = targets + (size_t)b * (T_INTERVALS * VW);
    const float* gO = outputs + (size_t)b * (T_INTERVALS * OW);

    // ---- Stage both rows into LDS via CDNA5 async global->LDS copies ----
    {
        const unsigned baseT = lds_byte_off(sT);
        const unsigned baseO = lds_byte_off(sO);
        // targets row: 640 x 16B chunks
        for (int i = tid; i < (T_INTERVALS * VW) / 4; i += TPB) {
            unsigned boff = (unsigned)i * 16u;
            unsigned laddr = baseT + boff;
            asm volatile("global_load_async_to_lds_b128 %0, %1, %2"
                         :: "v"(laddr), "v"(boff), "s"(gT) : "memory");
        }
        // outputs row: 512 x 16B chunks
        for (int i = tid; i < (T_INTERVALS * OW) / 4; i += TPB) {
            unsigned boff = (unsigned)i * 16u;
            unsigned laddr = baseO + boff;
            asm volatile("global_load_async_to_lds_b128 %0, %1, %2"
                         :: "v"(laddr), "v"(boff), "s"(gO) : "memory");
        }
        asm volatile("s_wait_asynccnt 0" ::: "memory");
    }
    __syncthreads();

    // ---- Per-thread: two timesteps each ----
    const int t0i = tid;
    const int t1i = tid + TPB;
    float lv0 = 0.0f, lv1 = 0.0f;
    int mt = -1;

    {
        const int t = t0i;
        float a0 = sT[t * VW + 0], a1 = sT[t * VW + 1], a2 = sT[t * VW + 2];
        float a3 = sT[t * VW + 3], a4 = sT[t * VW + 4];
        float o0 = sO[t * OW + 0], o1 = sO[t * OW + 1];
        float o2 = sO[t * OW + 2], o3 = sO[t * OW + 3];
        float per = (a0 + a1) + (a2 + a3) + a4;
        float censor = 1.0f - ((o0 + o1) + (o2 + o3));
        lv0 = a0 * __logf(censor + LOSS_EPS)
            + a1 * __logf(o0 + LOSS_EPS) + a2 * __logf(o1 + LOSS_EPS)
            + a3 * __logf(o2 + LOSS_EPS) + a4 * __logf(o3 + LOSS_EPS);
        if (per > 0.0f) mt = t;
    }
    {
        const int t = t1i;
        float a0 = sT[t * VW + 0], a1 = sT[t * VW + 1], a2 = sT[t * VW + 2];
        float a3 = sT[t * VW + 3], a4 = sT[t * VW + 4];
        float o0 = sO[t * OW + 0], o1 = sO[t * OW + 1];
        float o2 = sO[t * OW + 2], o3 = sO[t * OW + 3];
        float per = (a0 + a1) + (a2 + a3) + a4;
        float censor = 1.0f - ((o0 + o1) + (o2 + o3));
        lv1 = a0 * __logf(censor + LOSS_EPS)
            + a1 * __logf(o0 + LOSS_EPS) + a2 * __logf(o1 + LOSS_EPS)
            + a3 * __logf(o2 + LOSS_EPS) + a4 * __logf(o3 + LOSS_EPS);
        if (per > 0.0f) mt = t;   // t1i > t0i, so this dominates
    }

    // ---- Block max-reduce: last timestep with any positive target ----
    sI[tid] = mt;
    __syncthreads();
    for (int s = TPB / 2; s > 0; s >>= 1) {
        if (tid < s) sI[tid] = max(sI[tid], sI[tid + s]);
        __syncthreads();
    }
    const int tmax = sI[0];

    // ---- Block sum-reduce of masked loss ----
    float ls = 0.0f;
    if (t0i <= tmax) ls += lv0;
    if (t1i <= tmax) ls += lv1;
    sF[tid] = ls;
    __syncthreads();
    for (int s = TPB / 2; s > 0; s >>= 1) {
        if (tid < s) sF[tid] += sF[tid + s];
        __syncthreads();
    }

    if (tid == 0) {
        const float cnt = (float)(tmax + 1);
        if (use_atomic) {
            atomicAdd(&acc[0], sF[0]);
            atomicAdd(&acc[1], cnt);
        } else {
            acc[2 * b + 0] = sF[0];
            acc[2 * b + 1] = cnt;
        }
    }
}

// Deterministic fixed-order final reduction over per-block partials.
__global__ __launch_bounds__(TPB) void reduce_final(
    const float* __restrict__ partials, int nb, float* __restrict__ out)
{
    __shared__ float sL[TPB];
    __shared__ float sC[TPB];
    float l = 0.0f, c = 0.0f;
    for (int i = threadIdx.x; i < nb; i += TPB) {
        l += partials[2 * i + 0];
        c += partials[2 * i + 1];
    }
    sL[threadIdx.x] = l;
    sC[threadIdx.x] = c;
    __syncthreads();
    for (int s = TPB / 2; s > 0; s >>= 1) {
        if (threadIdx.x < s) {
            sL[threadIdx.x] += sL[threadIdx.x + s];
            sC[threadIdx.x] += sC[threadIdx.x + s];
        }
        __syncthreads();
    }
    if (threadIdx.x == 0) {
        float L = sL[0], C = sC[0];
        out[0] = (C > 0.0f) ? (-L / fmaxf(C, 1.0f)) : 0.0f;
    }
}

__global__ void zero2(float* p) {
    if (threadIdx.x < 2) p[threadIdx.x] = 0.0f;
}

extern "C" void kernel_launch(void* const* d_in, const int* in_sizes, int n_in,
                              void* d_out, int out_size, void* d_ws, size_t ws_size,
                              hipStream_t stream) {
    const float* outputs = (const float*)d_in[0];   // [B, T, 4] fp32
    const float* targets = (const float*)d_in[1];   // [B, T, 5] fp32
    float* out = (float*)d_out;                     // scalar fp32
    float* ws  = (float*)d_ws;

    const int B = in_sizes[1] / (T_INTERVALS * VW);
    const size_t need = (size_t)B * 2 * sizeof(float);

    if (ws_size >= need) {
        // Deterministic path: per-block partials + fixed-order reduction.
        censored_loss_rows<<<B, TPB, 0, stream>>>(outputs, targets, ws, 0);
        reduce_final<<<1, TPB, 0, stream>>>(ws, B, out);
    } else {
        // Fallback: atomic accumulation into ws[0..1].
        zero2<<<1, 64, 0, stream>>>(ws);
        censored_loss_rows<<<B, TPB, 0, stream>>>(outputs, targets, ws, 1);
        reduce_final<<<1, TPB, 0, stream>>>(ws, 1, out);
    }
}